// GraphConvolution_16801912062643
// MI455X (gfx1250) — compile-verified
//
#include <hip/hip_runtime.h>
#include <hip/hip_bf16.h>

typedef __attribute__((ext_vector_type(2))) float v2f;
typedef __attribute__((ext_vector_type(8))) float v8f;

#define D 128  // D_IN == D_OUT == 128

// -----------------------------------------------------------------------------
// Kernel 1: support = x @ W using V_WMMA_F32_16X16X4_F32.
// Block = 256 threads = 8 waves. Each wave computes a 16-row x 128-col strip.
// W is staged in LDS pair-interleaved along K so each B fragment (two K-rows,
// one column) is a single aligned ds_load_b64 straight into a WMMA src pair.
// -----------------------------------------------------------------------------
__global__ __launch_bounds__(256) void gcn_gemm_wmma(
    const float* __restrict__ x, const float* __restrict__ W,
    float* __restrict__ support, int n_nodes) {
  // Wlds[kk*128 + n] = { W[2kk][n], W[2kk+1][n] }  -> 64 KB of 320 KB WGP LDS
  __shared__ v2f Wlds[(D / 2) * D];

  const int tid = threadIdx.x;
  for (int p = tid; p < (D / 2) * D; p += 256) {
    const int kk = p >> 7;   // pair-row index (k/2)
    const int n  = p & 127;  // column
    v2f pr;
    pr.x = W[(2 * kk) * D + n];
    pr.y = W[(2 * kk + 1) * D + n];
    Wlds[p] = pr;
  }
  __syncthreads();

  const int wave = tid >> 5;
  const int lane = tid & 31;
  const int row0 = (blockIdx.x * 8 + wave) * 16;
  if (row0 >= n_nodes) return;  // wave-uniform exit: EXEC stays all-1s for WMMA

  const int m    = lane & 15;        // row (A) / col (B,C) within half-wave
  const int koff = (lane >> 4) << 1; // 0 for lanes 0-15, 2 for lanes 16-31

  int arow_idx = row0 + m;
  if (arow_idx >= n_nodes) arow_idx = n_nodes - 1;  // clamp tail loads
  const float* __restrict__ arow = x + (size_t)arow_idx * D;

  v8f acc[8] = {};  // 8 n-tiles of 16 cols -> full 128-col strip

  for (int k = 0; k < D; k += 4) {
    // A fragment: {x[row][k+koff], x[row][k+koff+1]} — 8B aligned (k%4==0, koff even)
    const v2f av = *(const v2f*)&arow[k + koff];
    // B fragments: pair index (k+koff)/2 * 128 + (t*16 + m)
    const v2f* __restrict__ bptr = &Wlds[((k + koff) >> 1) * D + m];
#pragma unroll
    for (int t = 0; t < 8; ++t) {
      const v2f bv = bptr[t * 16];  // ds_load_b64, immediate offset t*128B
      acc[t] = __builtin_amdgcn_wmma_f32_16x16x4_f32(
          /*neg_a=*/false, av, /*neg_b=*/false, bv,
          /*c_mod=*/(short)0, acc[t], /*reuse_a=*/false, /*reuse_b=*/false);
    }
  }

  // C/D layout: VGPR j -> M = row0 + j (+8 for lanes 16-31), N = t*16 + m
  const int mbase = row0 + ((lane >> 4) << 3);
#pragma unroll
  for (int j = 0; j < 8; ++j) {
    const int r = mbase + j;
    if (r < n_nodes) {
      float* __restrict__ orow = support + (size_t)r * D;
#pragma unroll
      for (int t = 0; t < 8; ++t) {
        orow[t * 16 + m] = acc[t][j];
      }
    }
  }
}

// -----------------------------------------------------------------------------
// Kernel 2: out[n][c] = bias[c]  (harness poisons d_out; we must re-init)
// -----------------------------------------------------------------------------
__global__ __launch_bounds__(256) void gcn_init_out(
    float* __restrict__ out, const float* __restrict__ bias, int total) {
  int i = blockIdx.x * blockDim.x + threadIdx.x;
  if (i < total) out[i] = bias[i & (D - 1)];
}

// -----------------------------------------------------------------------------
// Kernel 3: COO scatter SpMM. One wave per edge; each lane handles 4 floats.
// support (51.2 MB) fits in the 192 MB L2 -> gathers are L2-served.
// Edge metadata loads are forced scalar (wave-uniform e via readfirstlane).
// -----------------------------------------------------------------------------
__global__ __launch_bounds__(256) void gcn_spmm_scatter(
    const int* __restrict__ edge_src, const int* __restrict__ edge_dst,
    const float* __restrict__ edge_weight, const float* __restrict__ support,
    float* __restrict__ out, int n_edges) {
  // e is identical for all 32 lanes of a wave; make that explicit so the
  // compiler emits scalar (SMEM) loads for the per-edge metadata.
  const int e =
      __builtin_amdgcn_readfirstlane(blockIdx.x * 8 + (int)(threadIdx.x >> 5));
  if (e >= n_edges) return;
  const int lane = threadIdx.x & 31;
  const int c4 = lane * 4;

  const int s = edge_src[e];
  const int d = edge_dst[e];
  const float w = edge_weight[e];

  const float4 v = *(const float4*)&support[(size_t)s * D + c4];
  float* o = out + (size_t)d * D + c4;
  atomicAdd(o + 0, w * v.x);  // no-return -> global_atomic_add_f32 (STOREcnt)
  atomicAdd(o + 1, w * v.y);
  atomicAdd(o + 2, w * v.z);
  atomicAdd(o + 3, w * v.w);
}

// -----------------------------------------------------------------------------
// Launcher. Inputs (setup_inputs order):
//   0: x [N*128] f32, 1: edge_src [E] i32, 2: edge_dst [E] i32,
//   3: edge_weight [E] f32, 4: W [128*128] f32, 5: bias [128] f32
// d_out: [N*128] f32.  d_ws holds support [N*128] f32 (51.2 MB).
// -----------------------------------------------------------------------------
extern "C" void kernel_launch(void* const* d_in, const int* in_sizes, int n_in,
                              void* d_out, int out_size, void* d_ws, size_t ws_size,
                              hipStream_t stream) {
  const float* x           = (const float*)d_in[0];
  const int*   edge_src    = (const int*)d_in[1];
  const int*   edge_dst    = (const int*)d_in[2];
  const float* edge_weight = (const float*)d_in[3];
  const float* W           = (const float*)d_in[4];
  const float* bias        = (const float*)d_in[5];

  const int n_nodes = in_sizes[0] / D;
  const int n_edges = in_sizes[1];

  float* out     = (float*)d_out;
  float* support = (float*)d_ws;  // n_nodes * 128 floats

  // 1) support = x @ W  (WMMA f32)
  const int gemm_blocks = (n_nodes + 127) / 128;  // 128 rows/block (8 waves x 16)
  gcn_gemm_wmma<<<gemm_blocks, 256, 0, stream>>>(x, W, support, n_nodes);

  // 2) out = broadcast(bias)
  const int total = n_nodes * D;
  gcn_init_out<<<(total + 255) / 256, 256, 0, stream>>>(out, bias, total);

  // 3) out[dst] += w * support[src]   (one wave per edge)
  const int scatter_blocks = (n_edges + 7) / 8;  // 8 waves (edges) per block
  gcn_spmm_scatter<<<scatter_blocks, 256, 0, stream>>>(
      edge_src, edge_dst, edge_weight, support, out, n_edges);
}